// LocalPart_61460982006076
// MI455X (gfx1250) — compile-verified
//
#include <hip/hip_runtime.h>

// ---------------- problem constants ----------------
#define N_NODES 22
#define CH      64
#define NBT     3200            // B*T = 64*50
#define MROWS   70400           // NBT * N_NODES  (== 4400 * 16)
#define MS      4505600         // MROWS * CH
#define NEGV    (-9000000000000000.0f)
#define BN_EPSF 1e-5f

// ---------------- workspace layout (float offsets) ----------------
#define OFF_AOFF_S 0            // 64*22*22 = 30976
#define OFF_DIAG_S 30976        // 64*22    = 1408
#define OFF_AOFF_C 32384        // 30976
#define OFF_DIAG_C 63360        // 1408
#define OFF_SCALE  64768        // 4*64
#define OFF_SHIFT  65024        // 4*64
#define OFF_SUM    65280        // 4*64
#define OFF_SSQ    65536        // 4*64
#define OFF_BIG    65792        // 6 big buffers of MS floats follow

typedef __attribute__((ext_vector_type(2))) float v2f;
typedef __attribute__((ext_vector_type(8))) float v8f;

// ---- CDNA5 async global->LDS copy helpers (ASYNCcnt path, ISA 08) ----
// Generic pointers to __shared__ carry the LDS byte offset in addr[31:0]
// (flat aperture mapping), so truncation yields the VDST LDS address.
__device__ __forceinline__ unsigned lds_off_of(const void* p) {
    return (unsigned)(size_t)p;
}
__device__ __forceinline__ void async_ld_b128(unsigned ldsoff, const float* g) {
    asm volatile("global_load_async_to_lds_b128 %0, %1, off"
                 :: "v"(ldsoff), "v"(g) : "memory");
}
__device__ __forceinline__ void wait_async0() {
    asm volatile("s_wait_asynccnt 0x0" ::: "memory");
}

// ---------------------------------------------------------------
// 0) zero the BN statistic accumulators (harness poisons ws)
// ---------------------------------------------------------------
__global__ void zero_kernel(float* __restrict__ p, int n) {
    int i = blockIdx.x * 256 + threadIdx.x;
    if (i < n) p[i] = 0.0f;
}

// ---------------------------------------------------------------
// 1) build sparse-softmax adjacency: diag[o][i], aoff[o][i][j]
// ---------------------------------------------------------------
__global__ void graph_softmax(const float* __restrict__ e, const int* __restrict__ rows,
                              const int* __restrict__ cols, int nnz,
                              float* __restrict__ aoff, float* __restrict__ diag) {
    int r = blockIdx.x * blockDim.x + threadIdx.x;
    if (r >= CH * N_NODES) return;
    int o = r / N_NODES, i = r % N_NODES;
    float l[N_NODES];
#pragma unroll
    for (int j = 0; j < N_NODES; ++j) l[j] = NEGV;
    for (int k = 0; k < nnz; ++k)
        if (rows[k] == i) l[cols[k]] = e[o * nnz + k];
    float mx = l[0];
#pragma unroll
    for (int j = 1; j < N_NODES; ++j) mx = fmaxf(mx, l[j]);
    float s = 0.0f;
#pragma unroll
    for (int j = 0; j < N_NODES; ++j) { l[j] = __expf(l[j] - mx); s += l[j]; }
    float inv = 1.0f / s;
    diag[o * N_NODES + i] = l[i] * inv;
#pragma unroll
    for (int j = 0; j < N_NODES; ++j)
        aoff[(o * N_NODES + i) * N_NODES + j] = (j == i) ? 0.0f : l[j] * inv;
}

// ---------------------------------------------------------------
// 2) dis[bt][j][c] = sqrt(S2 - 2 x_j S1 + N x_j^2) + 1e-8  (closed form)
// ---------------------------------------------------------------
__global__ __launch_bounds__(64) void dis_kernel(const float* __restrict__ x,
                                                 float* __restrict__ dis) {
    int bt = blockIdx.x;
    int c  = threadIdx.x;
    const float* xb = x + (size_t)bt * N_NODES * CH + c;
    float s1 = 0.0f, s2 = 0.0f;
    float xv[N_NODES];
#pragma unroll
    for (int i = 0; i < N_NODES; ++i) { float v = xb[(size_t)i * CH]; xv[i] = v; s1 += v; s2 += v * v; }
    float* db = dis + (size_t)bt * N_NODES * CH + c;
#pragma unroll
    for (int j = 0; j < N_NODES; ++j) {
        float v = xv[j];
        float q = s2 - 2.0f * v * s1 + (float)N_NODES * v * v;
        db[(size_t)j * CH] = sqrtf(fmaxf(q, 0.0f)) + 1e-8f;
    }
}

// ---------------------------------------------------------------
// 3a) fused fp32 WMMA GEMM x4: one 16-row M-tile per block (128 thr = 4 waves),
//     A-tile async-staged to LDS once and shared by all 4 waves; each wave owns
//     a 16-col O stripe and accumulates against 4 weight matrices (4 WMMAs per
//     A-fragment load; x read once total instead of 4x).
// ---------------------------------------------------------------
__global__ __launch_bounds__(128) void gemm64x4_wmma(
    const float* __restrict__ X,
    const float* __restrict__ W0, const float* __restrict__ W1,
    const float* __restrict__ W2, const float* __restrict__ W3,
    float* __restrict__ Y0, float* __restrict__ Y1,
    float* __restrict__ Y2, float* __restrict__ Y3) {
    __shared__ __align__(16) float xt[16 * CH];     // 4 KB A-tile
    int tid  = threadIdx.x;
    int wave = tid >> 5, lane = tid & 31;
    int half = lane >> 4, lm = lane & 15;
    size_t rowbase = (size_t)blockIdx.x * 16;

    // stage X[rowbase..rowbase+16) x 64 into LDS: 256 x b128, 2 per thread
    {
        const float* g = X + rowbase * CH + tid * 4;
        unsigned     l = lds_off_of(&xt[tid * 4]);
        async_ld_b128(l, g);
        async_ld_b128(l + 2048, g + 512);
        wait_async0();
    }
    __syncthreads();

    int col = wave * 16 + lm;
    v8f a0 = {}, a1 = {}, a2 = {}, a3 = {};
#pragma unroll
    for (int k0 = 0; k0 < CH; k0 += 4) {
        int ka = k0 + 2 * half;
        v2f a; a.x = xt[lm * CH + ka]; a.y = xt[lm * CH + ka + 1];
        v2f b; b.x = W0[ka * CH + col]; b.y = W0[(ka + 1) * CH + col];
        a0 = __builtin_amdgcn_wmma_f32_16x16x4_f32(false, a, false, b, (short)0, a0, false, false);
        b.x = W1[ka * CH + col]; b.y = W1[(ka + 1) * CH + col];
        a1 = __builtin_amdgcn_wmma_f32_16x16x4_f32(false, a, false, b, (short)0, a1, false, false);
        b.x = W2[ka * CH + col]; b.y = W2[(ka + 1) * CH + col];
        a2 = __builtin_amdgcn_wmma_f32_16x16x4_f32(false, a, false, b, (short)0, a2, false, false);
        b.x = W3[ka * CH + col]; b.y = W3[(ka + 1) * CH + col];
        a3 = __builtin_amdgcn_wmma_f32_16x16x4_f32(false, a, false, b, (short)0, a3, false, false);
    }
    size_t obase = (rowbase + half * 8) * CH + col;
#pragma unroll
    for (int r = 0; r < 8; ++r) {
        Y0[obase + (size_t)r * CH] = a0[r];
        Y1[obase + (size_t)r * CH] = a1[r];
        Y2[obase + (size_t)r * CH] = a2[r];
        Y3[obase + (size_t)r * CH] = a3[r];
    }
}

// ---------------------------------------------------------------
// 3b) single-matrix variant (dis @ W_dis), same LDS-staged structure
// ---------------------------------------------------------------
__global__ __launch_bounds__(128) void gemm64_wmma_lds(const float* __restrict__ X,
                                                       const float* __restrict__ W,
                                                       float* __restrict__ Y) {
    __shared__ __align__(16) float xt[16 * CH];
    int tid  = threadIdx.x;
    int wave = tid >> 5, lane = tid & 31;
    int half = lane >> 4, lm = lane & 15;
    size_t rowbase = (size_t)blockIdx.x * 16;
    {
        const float* g = X + rowbase * CH + tid * 4;
        unsigned     l = lds_off_of(&xt[tid * 4]);
        async_ld_b128(l, g);
        async_ld_b128(l + 2048, g + 512);
        wait_async0();
    }
    __syncthreads();
    int col = wave * 16 + lm;
    v8f acc = {};
#pragma unroll
    for (int k0 = 0; k0 < CH; k0 += 4) {
        int ka = k0 + 2 * half;
        v2f a; a.x = xt[lm * CH + ka]; a.y = xt[lm * CH + ka + 1];
        v2f b; b.x = W[ka * CH + col]; b.y = W[(ka + 1) * CH + col];
        acc = __builtin_amdgcn_wmma_f32_16x16x4_f32(false, a, false, b, (short)0, acc, false, false);
    }
    size_t obase = (rowbase + half * 8) * CH + col;
#pragma unroll
    for (int r = 0; r < 8; ++r) Y[obase + (size_t)r * CH] = acc[r];
}

// ---------------------------------------------------------------
// 4) per-(b,t) mixing: xs = h0s*diag + Aoff@h1s ; same for con ;
//    z = att@zd + b_dis.  h1s/h1c/zd tiles staged global->LDS with
//    async b128 copies. Writes in place; accumulates BN stats.
// ---------------------------------------------------------------
__global__ __launch_bounds__(256) void mix_kernel(
    float* __restrict__ h0s, const float* __restrict__ h1s,
    float* __restrict__ h0c, const float* __restrict__ h1c,
    float* __restrict__ zd,
    const float* __restrict__ aoff_s, const float* __restrict__ diag_s,
    const float* __restrict__ aoff_c, const float* __restrict__ diag_c,
    const float* __restrict__ att, const float* __restrict__ b_dis,
    float* __restrict__ gsum, float* __restrict__ gssq) {
    __shared__ __align__(16) float l1s[N_NODES * CH];
    __shared__ __align__(16) float l1c[N_NODES * CH];
    __shared__ __align__(16) float lzd[N_NODES * CH];
    __shared__ float latt[N_NODES * N_NODES];
    __shared__ float ssum[3 * CH];
    __shared__ float sssq[3 * CH];
    int bt = blockIdx.x, tid = threadIdx.x;
    size_t base = (size_t)bt * N_NODES * CH;
    // async-stage the three 5632-byte tiles (352 b128 each)
    for (int i = tid; i < (N_NODES * CH) / 4; i += 256) {
        async_ld_b128(lds_off_of(&l1s[i * 4]), h1s + base + i * 4);
        async_ld_b128(lds_off_of(&l1c[i * 4]), h1c + base + i * 4);
        async_ld_b128(lds_off_of(&lzd[i * 4]), zd  + base + i * 4);
    }
    for (int i = tid; i < N_NODES * N_NODES; i += 256) latt[i] = att[i];
    for (int i = tid; i < 3 * CH; i += 256) { ssum[i] = 0.0f; sssq[i] = 0.0f; }
    wait_async0();
    __syncthreads();
    for (int i = tid; i < N_NODES * CH; i += 256) {
        int o = i & 63, n = i >> 6;
        float xsv = h0s[base + i] * diag_s[o * N_NODES + n];
        float ycv = h0c[base + i] * diag_c[o * N_NODES + n];
        float zv  = b_dis[o];
        const float* as = aoff_s + (o * N_NODES + n) * N_NODES;
        const float* ac = aoff_c + (o * N_NODES + n) * N_NODES;
        const float* at = latt + n * N_NODES;
#pragma unroll
        for (int j = 0; j < N_NODES; ++j) {
            xsv += as[j] * l1s[j * CH + o];
            ycv += ac[j] * l1c[j * CH + o];
            zv  += at[j] * lzd[j * CH + o];
        }
        h0s[base + i] = xsv;   // xs
        h0c[base + i] = ycv;   // yc
        zd[base + i]  = zv;    // z
        atomicAdd(&ssum[o], xsv);              atomicAdd(&sssq[o], xsv * xsv);
        atomicAdd(&ssum[CH + o], ycv);         atomicAdd(&sssq[CH + o], ycv * ycv);
        atomicAdd(&ssum[2 * CH + o], zv);      atomicAdd(&sssq[2 * CH + o], zv * zv);
    }
    __syncthreads();
    for (int i = tid; i < 3 * CH; i += 256) {
        atomicAdd(&gsum[i], ssum[i]);
        atomicAdd(&gssq[i], sssq[i]);
    }
}

// ---------------------------------------------------------------
// 5) BN finalize: scale = gamma*rsqrt(var+eps), shift = beta - mean*scale
// ---------------------------------------------------------------
__global__ void bn_finalize(const float* __restrict__ gsum, const float* __restrict__ gssq,
                            const float* __restrict__ gammas, const float* __restrict__ betas,
                            float* __restrict__ scale, float* __restrict__ shift,
                            int t0, int t1, float invcnt) {
    int o = threadIdx.x;
    if (o >= CH) return;
    for (int t = t0; t < t1; ++t) {
        float mean = gsum[t * CH + o] * invcnt;
        float var  = gssq[t * CH + o] * invcnt - mean * mean;
        float a    = gammas[t * CH + o] * rsqrtf(var + BN_EPSF);
        scale[t * CH + o] = a;
        shift[t * CH + o] = betas[t * CH + o] - mean * a;
    }
}

// ---------------------------------------------------------------
// 6) cat GEMM (K=192) with fused BN+ReLU on the A fragment:
//    outp[m][o2] = sum_f relu(scale[f]*src_f[m] + shift[f]) * cat_w[o2][f]
//    Accumulates out-BN stats (each lane's 8 outputs share one column).
// ---------------------------------------------------------------
__global__ __launch_bounds__(32) void catgemm_wmma(
    const float* __restrict__ xs, const float* __restrict__ yc, const float* __restrict__ z,
    const float* __restrict__ cat_w, const float* __restrict__ scale,
    const float* __restrict__ shift, float* __restrict__ outp,
    float* __restrict__ gsum, float* __restrict__ gssq) {
    int lane = threadIdx.x;
    int half = lane >> 4, lm = lane & 15;
    int row = blockIdx.x * 16 + lm;
    int col = blockIdx.y * 16 + lm;
    v8f acc = {};
#pragma unroll
    for (int k0 = 0; k0 < 3 * CH; k0 += 4) {
        int f0 = k0 + 2 * half;
        v2f a, b;
#pragma unroll
        for (int u = 0; u < 2; ++u) {
            int f = f0 + u;
            const float* src = (f < CH) ? xs : (f < 2 * CH) ? yc : z;
            float v = src[(size_t)row * CH + (f & 63)];
            v = fmaxf(v * scale[f] + shift[f], 0.0f);
            if (u == 0) a.x = v; else a.y = v;
        }
        b.x = cat_w[col * (3 * CH) + f0];
        b.y = cat_w[col * (3 * CH) + f0 + 1];
        acc = __builtin_amdgcn_wmma_f32_16x16x4_f32(false, a, false, b,
                                                    (short)0, acc, false, false);
    }
    size_t obase = (size_t)(blockIdx.x * 16 + half * 8) * CH + col;
    float s = 0.0f, q = 0.0f;
#pragma unroll
    for (int r = 0; r < 8; ++r) {
        float v = acc[r];
        outp[obase + (size_t)r * CH] = v;
        s += v; q += v * v;
    }
    atomicAdd(&gsum[3 * CH + col], s);
    atomicAdd(&gssq[3 * CH + col], q);
}

// ---------------------------------------------------------------
// 7) final BN + ReLU -> d_out ((B,T,N,O) row-major == (M,O))
// ---------------------------------------------------------------
__global__ void final_kernel(const float* __restrict__ outp, const float* __restrict__ scale,
                             const float* __restrict__ shift, float* __restrict__ out) {
    int i = blockIdx.x * 256 + threadIdx.x;
    int o = i & 63;
    out[i] = fmaxf(outp[i] * scale[3 * CH + o] + shift[3 * CH + o], 0.0f);
}

// ---------------------------------------------------------------
extern "C" void kernel_launch(void* const* d_in, const int* in_sizes, int n_in,
                              void* d_out, int out_size, void* d_ws, size_t ws_size,
                              hipStream_t stream) {
    (void)n_in; (void)out_size; (void)ws_size;
    const float* x      = (const float*)d_in[0];
    const float* W_sym  = (const float*)d_in[1];
    const float* e_sym  = (const float*)d_in[2];
    const float* W_con  = (const float*)d_in[3];
    const float* e_con  = (const float*)d_in[4];
    const float* W_dis  = (const float*)d_in[5];
    const float* att    = (const float*)d_in[6];
    const float* b_dis  = (const float*)d_in[7];
    const float* cat_w  = (const float*)d_in[8];
    const float* gammas = (const float*)d_in[9];
    const float* betas  = (const float*)d_in[10];
    const int*   rows_s = (const int*)d_in[11];
    const int*   cols_s = (const int*)d_in[12];
    const int*   rows_c = (const int*)d_in[13];
    const int*   cols_c = (const int*)d_in[14];
    int nnz_s = in_sizes[11];
    int nnz_c = in_sizes[13];

    float* w      = (float*)d_ws;
    float* aoff_s = w + OFF_AOFF_S;
    float* diag_s = w + OFF_DIAG_S;
    float* aoff_c = w + OFF_AOFF_C;
    float* diag_c = w + OFF_DIAG_C;
    float* scale  = w + OFF_SCALE;
    float* shift  = w + OFF_SHIFT;
    float* gsum   = w + OFF_SUM;
    float* gssq   = w + OFF_SSQ;
    float* H0S = w + OFF_BIG;            // -> xs after mix
    float* H1S = H0S + MS;
    float* H0C = H1S + MS;               // -> yc after mix
    float* H1C = H0C + MS;
    float* ZD  = H1C + MS;               // -> z after mix
    float* DIS = ZD + MS;                // dis, then reused as pre-BN output

    zero_kernel<<<2, 256, 0, stream>>>(gsum, 512);   // SUM + SSQ contiguous

    graph_softmax<<<6, 256, 0, stream>>>(e_sym, rows_s, cols_s, nnz_s, aoff_s, diag_s);
    graph_softmax<<<6, 256, 0, stream>>>(e_con, rows_c, cols_c, nnz_c, aoff_c, diag_c);

    dis_kernel<<<NBT, 64, 0, stream>>>(x, DIS);

    // fused 4-way x GEMM: x read once, 4 WMMA accumulators
    gemm64x4_wmma<<<MROWS / 16, 128, 0, stream>>>(x,
        W_sym, W_sym + CH * CH, W_con, W_con + CH * CH,
        H0S, H1S, H0C, H1C);

    gemm64_wmma_lds<<<MROWS / 16, 128, 0, stream>>>(DIS, W_dis, ZD);

    mix_kernel<<<NBT, 256, 0, stream>>>(H0S, H1S, H0C, H1C, ZD,
                                        aoff_s, diag_s, aoff_c, diag_c,
                                        att, b_dis, gsum, gssq);

    bn_finalize<<<1, 64, 0, stream>>>(gsum, gssq, gammas, betas, scale, shift,
                                      0, 3, 1.0f / (float)MROWS);

    dim3 g(MROWS / 16, CH / 16);
    catgemm_wmma<<<g, 32, 0, stream>>>(H0S, H0C, ZD, cat_w, scale, shift,
                                       DIS, gsum, gssq);

    bn_finalize<<<1, 64, 0, stream>>>(gsum, gssq, gammas, betas, scale, shift,
                                      3, 4, 1.0f / (float)MROWS);

    final_kernel<<<(MROWS * CH) / 256, 256, 0, stream>>>(DIS, scale, shift, (float*)d_out);
}